// NPChangeTransitionPrior_14353780703782
// MI455X (gfx1250) — compile-verified
//
#include <hip/hip_runtime.h>
#include <hip/hip_bf16.h>
#include <math.h>

// ---------------------------------------------------------------------------
// NPChangeTransitionPrior: Z=8 per-latent MLPs (33->64->64->1, LeakyReLU) + JVP
// wrt last input coord, fused into one kernel using CDNA5 f32 WMMA
// (v_wmma_f32_16x16x4_f32) so the matrix pipes do the GEMM work at full fp32
// precision (matches the f32 reference bit-for-bit math class).
// ---------------------------------------------------------------------------

#define ZC    8
#define EC    32
#define HC    64
#define BC    64
#define LLC   512
#define LAGSC 2
#define LENC  (LLC - LAGSC)        // 510
#define NN    (BC * LENC)          // 32640
#define RES_SIZE (NN * ZC)         // 261120
#define LDSPAD 68                  // 64 + 4 floats row stride (bank-conflict pad)

typedef __attribute__((ext_vector_type(2))) float v2f;
typedef __attribute__((ext_vector_type(8))) float v8f;

static __device__ __forceinline__ v8f wmma_f32(v2f a, v2f b, v8f c) {
    // D = A(16x4 f32) * B(4x16 f32) + C(16x16 f32)
    return __builtin_amdgcn_wmma_f32_16x16x4_f32(
        /*neg_a=*/false, a, /*neg_b=*/false, b,
        /*c_mod=*/(short)0, c, /*reuse_a=*/false, /*reuse_b=*/false);
}

__global__ __launch_bounds__(128)
void npctp_mlp_jvp_kernel(const float* __restrict__ x,
                          const float* __restrict__ emb,
                          const float* __restrict__ W1,
                          const float* __restrict__ b1,
                          const float* __restrict__ W2,
                          const float* __restrict__ b2,
                          const float* __restrict__ W3,
                          const float* __restrict__ b3,
                          float* __restrict__ out)
{
    // Per-wave re-layout buffers (C-layout tile -> A-fragments), padded rows.
    __shared__ float buf_h[4][16][LDSPAD];   // h1 tile per wave
    __shared__ float buf_d[4][16][LDSPAD];   // dh1 tile per wave
    __shared__ float lds_ld[16][ZC];         // log|dout| per (row, z), fixed-order sum

    const int tid  = threadIdx.x;
    const int wave = tid >> 5;
    const int lane = tid & 31;
    const int hi   = lane >> 4;    // lane-half: 0 or 1
    const int lo   = lane & 15;
    const int n0   = blockIdx.x * 16;   // 16 samples per block, 2040 blocks

    // ---- A-fragments of the embedding tile: shared across both z's this wave does
    // A 16x4 layout: row = lane%16; a.x holds K = 2*hi, a.y holds K = 2*hi+1
    v2f aE[8];
    {
        const int n  = n0 + lo;
        const int b  = n / LENC;
        const int li = n - b * LENC;
        const float* ep = emb + (size_t)(b * LLC + li + LAGSC) * EC;
        #pragma unroll
        for (int kk = 0; kk < 8; ++kk) {
            const int e = 4 * kk + 2 * hi;
            aE[kk].x = ep[e];
            aE[kk].y = ep[e + 1];
        }
    }

    #pragma unroll
    for (int zi = 0; zi < 2; ++zi) {
        const int z = wave + 4 * zi;

        // x_t values matching this lane's C-layout rows (m = j + 8*hi)
        float xt[8];
        #pragma unroll
        for (int j = 0; j < 8; ++j) {
            const int n  = n0 + j + 8 * hi;
            const int b  = n / LENC;
            const int li = n - b * LENC;
            xt[j] = x[(size_t)(b * LLC + li + LAGSC) * ZC + z];
        }

        const float* W1z = W1 + z * (EC + 1) * HC;
        const float* W2z = W2 + z * HC * HC;
        const float* b1z = b1 + z * HC;
        const float* b2z = b2 + z * HC;
        const float* W3z = W3 + z * HC;
        const float  b3z = b3[z];

        // ---------------- Layer 1: pre1 = emb@W1[:32] + x_t*W1[32] + b1 ------
        #pragma unroll
        for (int c = 0; c < 4; ++c) {
            const int col = 16 * c + lo;
            v8f acc = {0.f,0.f,0.f,0.f,0.f,0.f,0.f,0.f};
            #pragma unroll
            for (int kk = 0; kk < 8; ++kk) {
                const int k = 4 * kk + 2 * hi;     // B 4x16: b.x row k, b.y row k+1
                v2f bf;
                bf.x = W1z[(size_t)k * HC + col];
                bf.y = W1z[(size_t)(k + 1) * HC + col];
                acc = wmma_f32(aE[kk], bf, acc);
            }
            const float wlast = W1z[(size_t)EC * HC + col];  // d(pre1)/d(x_t)
            const float bb    = b1z[col];
            #pragma unroll
            for (int j = 0; j < 8; ++j) {
                const float pre = acc[j] + xt[j] * wlast + bb;
                const float f   = (pre >= 0.f) ? 1.f : 0.01f;   // leaky'
                buf_h[wave][j + 8 * hi][col] = f * pre;          // h1
                buf_d[wave][j + 8 * hi][col] = f * wlast;        // dh1 (JVP)
            }
        }

        // Re-layout via LDS into layer-2 A-fragments (same-wave DS ops, in order)
        v2f ah[16], ad[16];
        #pragma unroll
        for (int kk = 0; kk < 16; ++kk) {
            const int k = 4 * kk + 2 * hi;
            ah[kk].x = buf_h[wave][lo][k];
            ah[kk].y = buf_h[wave][lo][k + 1];
            ad[kk].x = buf_d[wave][lo][k];
            ad[kk].y = buf_d[wave][lo][k + 1];
        }

        // ---------------- Layer 2 (value + tangent share W2) + layer 3 -------
        float pout[8] = {0,0,0,0,0,0,0,0};
        float pdt [8] = {0,0,0,0,0,0,0,0};
        #pragma unroll
        for (int c = 0; c < 4; ++c) {
            const int col = 16 * c + lo;
            v8f acc = {0.f,0.f,0.f,0.f,0.f,0.f,0.f,0.f};
            v8f dac = {0.f,0.f,0.f,0.f,0.f,0.f,0.f,0.f};
            #pragma unroll
            for (int kk = 0; kk < 16; ++kk) {
                const int k = 4 * kk + 2 * hi;
                v2f bf;
                bf.x = W2z[(size_t)k * HC + col];
                bf.y = W2z[(size_t)(k + 1) * HC + col];
                acc = wmma_f32(ah[kk], bf, acc);
                dac = wmma_f32(ad[kk], bf, dac);
            }
            const float bb  = b2z[col];
            const float w3v = W3z[col];
            #pragma unroll
            for (int j = 0; j < 8; ++j) {
                const float pre = acc[j] + bb;
                const float f   = (pre >= 0.f) ? 1.f : 0.01f;
                pout[j] += (f * pre)    * w3v;   // h2 . W3
                pdt [j] += (f * dac[j]) * w3v;   // dh2 . W3
            }
        }

        // Reduce the hidden dim across the 16 lanes of each lane-half.
        #pragma unroll
        for (int j = 0; j < 8; ++j) {
            float a = pout[j], d = pdt[j];
            #pragma unroll
            for (int off = 1; off < 16; off <<= 1) {
                a += __shfl_xor(a, off, 32);
                d += __shfl_xor(d, off, 32);
            }
            pout[j] = a; pdt[j] = d;
        }

        if (lo == 0) {
            #pragma unroll
            for (int j = 0; j < 8; ++j) {
                const int m = j + 8 * hi;
                const int n = n0 + m;
                out[(size_t)n * ZC + z] = pout[j] + b3z;          // residuals
                lds_ld[m][z] = __logf(fabsf(pdt[j]));             // per-z logdet
            }
        }
    }

    __syncthreads();
    if (tid < 16) {
        float s = 0.f;
        #pragma unroll
        for (int z = 0; z < ZC; ++z) s += lds_ld[tid][z];         // fixed z order
        out[RES_SIZE + n0 + tid] = s;                             // log_abs_det
    }
}

extern "C" void kernel_launch(void* const* d_in, const int* in_sizes, int n_in,
                              void* d_out, int out_size, void* d_ws, size_t ws_size,
                              hipStream_t stream) {
    (void)in_sizes; (void)n_in; (void)out_size; (void)d_ws; (void)ws_size;
    const float* x   = (const float*)d_in[0];
    const float* emb = (const float*)d_in[1];
    const float* W1  = (const float*)d_in[2];
    const float* b1  = (const float*)d_in[3];
    const float* W2  = (const float*)d_in[4];
    const float* b2  = (const float*)d_in[5];
    const float* W3  = (const float*)d_in[6];
    const float* b3  = (const float*)d_in[7];
    float* out = (float*)d_out;

    dim3 grid(NN / 16);   // 2040 blocks x 16 samples
    dim3 block(128);      // 4 wave32s; each wave handles 2 latents
    npctp_mlp_jvp_kernel<<<grid, block, 0, stream>>>(x, emb, W1, b1, W2, b2,
                                                     W3, b3, out);
}